// HMQSoftmax_59983513256151
// MI455X (gfx1250) — compile-verified
//
#include <hip/hip_runtime.h>
#include <cstdint>

typedef float v4f __attribute__((ext_vector_type(4)));

#define ROW_LEN    2048
#define WPB        4                    // waves per block (blockDim = 128)
#define BUF_BYTES  (ROW_LEN * 4)        // 8 KB per row buffer

#define LN2F       0.69314718055994530942f
#define INV_LN2F   1.44269504088896340736f

// q = floor(x / ln2): mul + 2-fma Newton-refined quotient (~correctly rounded divide),
// fp_to_sint(floorf(y)) folds to v_cvt_flr_i32_f32; 2^q exact via v_ldexp_f32.
__device__ __forceinline__ float pow2_floor(float xi) {
  float y = xi * INV_LN2F;
  float r = __builtin_fmaf(-LN2F, y, xi);      // residual x - ln2*y (exact-ish via fma)
  y = __builtin_fmaf(r, INV_LN2F, y);          // refined quotient, ~0.5 ulp
  int qi = (int)floorf(y);                     // v_cvt_flr_i32_f32
  return ldexpf(1.0f, qi);                     // v_ldexp_f32, bit-exact 2^q
}

__global__ __launch_bounds__(WPB * 32, 1)
void hmq_softmax_bfp(const float* __restrict__ x, float* __restrict__ out, int nrows) {
  extern __shared__ char smem[];
  const int lane = threadIdx.x & 31;
  const int wave = threadIdx.x >> 5;
  const int gw = blockIdx.x * WPB + wave;
  const int gstride = gridDim.x * WPB;

  char* bufA = smem + (wave * 2 + 0) * BUF_BYTES;
  char* bufB = smem + (wave * 2 + 1) * BUF_BYTES;

  if (gw >= nrows) return;

  // Async DMA one 8KB row into an LDS buffer: 16 x b128 per wave (ASYNCcnt += 16).
  auto prefetch = [&](int row, char* buf) {
    const char* g = (const char*)(x + (size_t)row * ROW_LEN) + lane * 16;
    char* l = buf + lane * 16;
#pragma unroll
    for (int j = 0; j < 16; ++j) {
      unsigned lds_addr = (unsigned)(uintptr_t)(l + j * 512); // low 32b of flat = LDS offset
      unsigned long long ga = (unsigned long long)(uintptr_t)(g + j * 512);
      asm volatile("global_load_async_to_lds_b128 %0, %1, off"
                   :: "v"(lds_addr), "v"(ga) : "memory");
    }
  };

  int row = gw;
  int cur = 0;
  prefetch(row, bufA);

  for (;;) {
    const int nxt = row + gstride;
    const bool has_next = (nxt < nrows);
    if (has_next) {
      prefetch(nxt, cur ? bufA : bufB);                       // next row -> other buffer
      asm volatile("s_wait_asynccnt 0x10" ::: "memory");      // current row's 16 retired
    } else {
      asm volatile("s_wait_asynccnt 0x0" ::: "memory");
    }

    v4f* sv = (v4f*)(cur ? bufB : bufA);

    // ---- pass 1: exp_x = 2^floor(x/ln2); stash exp_x in-place in LDS; row sum ----
    float s = 0.0f;
#pragma unroll
    for (int j = 0; j < 16; ++j) {
      v4f v = sv[j * 32 + lane];
      v4f e;
      e.x = pow2_floor(v.x);
      e.y = pow2_floor(v.y);
      e.z = pow2_floor(v.z);
      e.w = pow2_floor(v.w);
      sv[j * 32 + lane] = e;                 // overwrite x with exp_x (same wave, DS in-order)
      s += (e.x + e.y) + (e.z + e.w);
    }
#pragma unroll
    for (int m = 16; m >= 1; m >>= 1) s += __shfl_xor(s, m, 32);

    // ---- Quake rsqrt on bf16 bits (RNE f32->bf16, magic 24375, 1 Newton step) ----
    unsigned u = __float_as_uint(s);
    unsigned rnd = 0x7FFFu + ((u >> 16) & 1u);
    short i16 = (short)(unsigned short)((u + rnd) >> 16);
    i16 = (short)(24375 - (i16 >> 1));
    float yq = __uint_as_float(((unsigned)(unsigned short)i16) << 16);
    float rs = yq * (1.5f - ((0.5f * s) * yq) * yq);

    // ---- pass 2: d = (exp_x*rs)*rs; block-16 BFP quant (block = one lane-quad) ----
    v4f* ov = (v4f*)(out + (size_t)row * ROW_LEN);
#pragma unroll
    for (int j = 0; j < 16; ++j) {
      v4f e = sv[j * 32 + lane];
      v4f d;
      d.x = (e.x * rs) * rs;
      d.y = (e.y * rs) * rs;
      d.z = (e.z * rs) * rs;
      d.w = (e.w * rs) * rs;

      float m = fmaxf(fmaxf(fabsf(d.x), fabsf(d.y)), fmaxf(fabsf(d.z), fabsf(d.w)));
      m = fmaxf(m, __shfl_xor(m, 1, 32));
      m = fmaxf(m, __shfl_xor(m, 2, 32));

      // shared_exp = floor(log2(m)) + 1 == frexp exponent (exact, incl. powers of 2)
      int ex = 0;
      (void)frexpf(m, &ex);
      float scale = ldexpf(1.0f, ex - 7);    // exact powers of two
      float invs  = ldexpf(1.0f, 7 - ex);

      v4f q;  // round half-to-even like jnp.round, clamp [-128,127] via v_med3
      q.x = __builtin_amdgcn_fmed3f(rintf(d.x * invs), -128.0f, 127.0f) * scale;
      q.y = __builtin_amdgcn_fmed3f(rintf(d.y * invs), -128.0f, 127.0f) * scale;
      q.z = __builtin_amdgcn_fmed3f(rintf(d.z * invs), -128.0f, 127.0f) * scale;
      q.w = __builtin_amdgcn_fmed3f(rintf(d.w * invs), -128.0f, 127.0f) * scale;

      __builtin_nontemporal_store(q, ov + j * 32 + lane);   // streaming b128 store
    }

    if (!has_next) break;
    row = nxt;
    cur ^= 1;
  }
}

extern "C" void kernel_launch(void* const* d_in, const int* in_sizes, int n_in,
                              void* d_out, int out_size, void* d_ws, size_t ws_size,
                              hipStream_t stream) {
  const float* x = (const float*)d_in[0];
  float* out = (float*)d_out;
  const int nrows = in_sizes[0] / ROW_LEN;      // 32768 for [1,16,2048,2048]
  int blocks = (nrows + WPB - 1) / WPB;
  if (blocks > 2048) blocks = 2048;             // grid-stride: 4 rows/wave at full size
  const size_t shmem = (size_t)WPB * 2 * BUF_BYTES;  // 64 KB (double-buffered per wave)
  hmq_softmax_bfp<<<blocks, WPB * 32, shmem, stream>>>(x, out, nrows);
}